// EsmFoldSelfAttention_86827058856666
// MI455X (gfx1250) — compile-verified
//
#include <hip/hip_runtime.h>
#include <hip/hip_bf16.h>
#include <stdint.h>

typedef __attribute__((ext_vector_type(2))) float v2f;
typedef __attribute__((ext_vector_type(8))) float v8f;
typedef __attribute__((ext_vector_type(4))) unsigned int v4u;
typedef __attribute__((ext_vector_type(4))) int v4i;
typedef __attribute__((ext_vector_type(8))) int v8i;

#define WMMA_F32(a, b, c) \
  __builtin_amdgcn_wmma_f32_16x16x4_f32(false, (a), false, (b), (short)0, (c), false, false)

namespace esm {
constexpr int B = 2, L = 1024, C = 1024, H = 32, D = 32;
constexpr float QSCALE = 0.17677669529663687f;  // 1/sqrt(32)
}

// ---------------------------------------------------------------------------
// TDM: issue a 2D fp32 tile load (tile_w x tile_h, row stride in elements)
// into LDS at byte offset lds_off, padding +4 dwords after every 32-dword row
// so the LDS image has a 36-float row stride. D# layout per ISA 08 §8.3/8.4.
// This toolchain exposes the 6-arg builtin:
//   (uint32x4 g0, int32x8 g1, int32x4 g2, int32x4 g3, int32x8 extra, i32 cpol)
// ---------------------------------------------------------------------------
__device__ __forceinline__ void tdm_load_tile_f32(
    uint32_t lds_off, const float* gptr, uint32_t tile_w, uint32_t tile_h,
    uint64_t row_stride_elems) {
  const uint64_t ga = (uint64_t)(uintptr_t)gptr;
  v4u g0;
  g0[0] = 1u;                                    // count=1 (valid user D#)
  g0[1] = lds_off;                               // lds_addr (bytes)
  g0[2] = (uint32_t)(ga & 0xffffffffu);          // global_addr[31:0]
  g0[3] = (uint32_t)((ga >> 32) & 0x1ffffffu)    // global_addr[56:32]
          | (2u << 30);                          // type = 2 ("image")
  v8i g1;
  // data_size=4B (2), pad_enable=1, pad_interval: 1<<4 qwords = 32 dwords,
  // pad_amount: field 3 -> 4 dwords
  g1[0] = (int)((2u << 16) | (1u << 20) | (4u << 22) | (3u << 25));
  g1[1] = (int)((tile_w & 0xffffu) << 16);                       // tensor_dim0 lo
  g1[2] = (int)(((tile_w >> 16) & 0xffffu) |
                ((tile_h & 0xffffu) << 16));                     // dim0 hi | dim1 lo
  g1[3] = (int)(((tile_h >> 16) & 0xffffu) |
                ((tile_w & 0xffffu) << 16));                     // dim1 hi | tile_dim0
  g1[4] = (int)(tile_h & 0xffffu);                               // tile_dim1 (dim2=0)
  g1[5] = (int)(uint32_t)(row_stride_elems & 0xffffffffu);       // dim0_stride lo
  g1[6] = (int)(uint32_t)((row_stride_elems >> 32) & 0xffffu);   // dim0_stride hi
  g1[7] = 0;                                                     // dim1_stride = 0
  const v4i z4 = {0, 0, 0, 0};                     // 2D: groups 2/3 unused
  const v8i z8 = {0, 0, 0, 0, 0, 0, 0, 0};
  __builtin_amdgcn_tensor_load_to_lds(g0, g1, z4, z4, z8, 0);
}

// ---------------------------------------------------------------------------
// Generic fp32 WMMA GEMM:  Out[M,N] = A[M,K] @ W[N,K]^T  (+ mode epilogue)
// TDM double-buffered K-tile staging: DMA of tile i+1 overlaps WMMA of tile i.
// MODE 0: QKV scatter to q/k/v [B,H,L,D], q scaled.
// MODE 1: gated = sigmoid(acc + b_g[n]) * y_ws[m,n]
// MODE 2: out   = acc + b_o[n]
// ---------------------------------------------------------------------------
template <int MODE>
__global__ __launch_bounds__(256) void gemm_wmma_f32(
    const float* __restrict__ A, const float* __restrict__ W,
    const float* __restrict__ bias1, const float* __restrict__ mul,
    float* __restrict__ o0, float* __restrict__ o1, float* __restrict__ o2,
    int M, int N, int K) {
  using namespace esm;
  const int tid = threadIdx.x;
  const int wave = tid >> 5;
  const int lane = tid & 31;
  const int hi = lane >> 4;
  const int nl = lane & 15;
  const int tileM = blockIdx.y * 64;
  const int tileN = blockIdx.x * 64;

  // 36-float row stride: 144B rows (16B aligned), even stride -> b64-clean
  __shared__ float As[2][64][36];
  __shared__ float Ws[2][64][36];
  constexpr uint32_t BUFBYTES = 64 * 36 * 4;
  const uint32_t asBase = (uint32_t)(uintptr_t)&As[0][0][0];  // LDS offset = addr[31:0]
  const uint32_t wsBase = (uint32_t)(uintptr_t)&Ws[0][0][0];

  const int rowblk = wave & 3;   // 16-row slice of the 64-row tile
  const int colpair = wave >> 2; // 32-col slice (two 16-wide accumulators)

  v8f acc0 = {0.f, 0.f, 0.f, 0.f, 0.f, 0.f, 0.f, 0.f};
  v8f acc1 = {0.f, 0.f, 0.f, 0.f, 0.f, 0.f, 0.f, 0.f};

  const float* aTile = A + (size_t)tileM * K;
  const float* wTile = W + (size_t)tileN * K;
  const int nk = K / 32;

  if (wave == 0) {  // prime buffer 0 (EXEC ignored by TDM; one issue per wave)
    tdm_load_tile_f32(asBase, aTile, 32, 64, (uint64_t)K);
    tdm_load_tile_f32(wsBase, wTile, 32, 64, (uint64_t)K);
  }

  for (int i = 0; i < nk; ++i) {
    const int buf = i & 1;
    if (wave == 0) {
      if (i + 1 < nk) {  // kick off next tile into the other buffer
        const uint32_t nb = (uint32_t)((i + 1) & 1) * BUFBYTES;
        tdm_load_tile_f32(asBase + nb, aTile + (i + 1) * 32, 32, 64, (uint64_t)K);
        tdm_load_tile_f32(wsBase + nb, wTile + (i + 1) * 32, 32, 64, (uint64_t)K);
        __builtin_amdgcn_s_wait_tensorcnt(2);  // retire oldest pair (tile i)
      } else {
        __builtin_amdgcn_s_wait_tensorcnt(0);
      }
    }
    __syncthreads();  // publish TDM-written LDS to all waves

    const int mrow = rowblk * 16 + nl;       // A-frag: M = lane&15
    const int ncol = colpair * 32 + nl;      // B-frag: N = lane&15
#pragma unroll
    for (int kk = 0; kk < 32; kk += 4) {
      v2f af = *(const v2f*)&As[buf][mrow][kk + 2 * hi];
      v2f b0 = *(const v2f*)&Ws[buf][ncol][kk + 2 * hi];
      v2f b1 = *(const v2f*)&Ws[buf][ncol + 16][kk + 2 * hi];
      acc0 = WMMA_F32(af, b0, acc0);
      acc1 = WMMA_F32(af, b1, acc1);
    }
    __syncthreads();  // all reads done before tile i+2 overwrites this buffer
  }

  // Epilogue. C/D layout: lane<16 -> (M=v, N=lane); lane>=16 -> (M=v+8, N=lane-16)
#pragma unroll
  for (int v = 0; v < 8; ++v) {
    const int m = tileM + rowblk * 16 + hi * 8 + v;
#pragma unroll
    for (int t = 0; t < 2; ++t) {
      const int n = tileN + colpair * 32 + nl + t * 16;
      const float c = (t == 0) ? acc0[v] : acc1[v];
      if (MODE == 0) {
        const int h = n / 96, rr = n % 96;
        const int part = rr >> 5, d = rr & 31;
        const int b = m >> 10, l = m & (L - 1);
        const size_t idx = (((size_t)(b * H + h)) * L + l) * D + d;
        if (part == 0)      o0[idx] = c * QSCALE;
        else if (part == 1) o1[idx] = c;
        else                o2[idx] = c;
      } else if (MODE == 1) {
        const float g = 1.0f / (1.0f + __expf(-(c + bias1[n])));
        o0[(size_t)m * N + n] = g * mul[(size_t)m * N + n];
      } else {
        o0[(size_t)m * N + n] = c + bias1[n];
      }
    }
  }
}

// ---------------------------------------------------------------------------
// Attention: per (b,h,32-query block). Two-pass online softmax, full prob
// matrix written to a_out [B,Lk,H,Lq]; y accumulated via WMMA P@V.
// ---------------------------------------------------------------------------
__device__ __forceinline__ void score_chunk(
    const float (*Qs)[36], float (*S)[66], const float* __restrict__ kptr,
    const float* __restrict__ bias, const unsigned char* __restrict__ mask,
    int b, int h, int qbase, int kc, int wave, int lane) {
  using namespace esm;
  const int qs = wave & 1;     // 16-row query slice
  const int lkb = wave >> 1;   // 16-col key slice (4 of them = 64-key chunk)
  const int hi = lane >> 4;
  const int nl = lane & 15;

  v8f acc = {0.f, 0.f, 0.f, 0.f, 0.f, 0.f, 0.f, 0.f};
  const int mrow = qs * 16 + nl;
  const float* krow = kptr + (size_t)(kc + lkb * 16 + nl) * D;
#pragma unroll
  for (int dd = 0; dd < 32; dd += 4) {
    v2f af = *(const v2f*)&Qs[mrow][dd + 2 * hi];
    v2f bf = *(const v2f*)&krow[dd + 2 * hi];   // k row contiguous in d -> b64 load
    acc = WMMA_F32(af, bf, acc);
  }

  const int lkg = kc + lkb * 16 + nl;
  const bool valid = mask[b * L + lkg] != 0;
#pragma unroll
  for (int vv = 0; vv < 8; ++vv) {
    const int lq = qs * 16 + vv + 8 * hi;
    const float bv = bias[(((size_t)(b * L + qbase + lq)) * L + lkg) * H + h];
    const float x = acc[vv] + bv;
    S[lq][lkb * 16 + nl] = valid ? x : -__builtin_inff();
  }
}

__global__ __launch_bounds__(256) void attn_wmma_f32(
    const float* __restrict__ q, const float* __restrict__ k,
    const float* __restrict__ v, const float* __restrict__ bias,
    const unsigned char* __restrict__ mask,
    float* __restrict__ a_out, float* __restrict__ y_ws) {
  using namespace esm;
  const int h = blockIdx.x, qb = blockIdx.y, b = blockIdx.z;
  const int bh = b * H + h;
  const int qbase = qb * 32;
  const int tid = threadIdx.x, wave = tid >> 5, lane = tid & 31;
  const int hi = lane >> 4, nl = lane & 15;

  __shared__ float Qs[32][36];
  __shared__ float S[32][66];          // 64-key score/prob chunk, even stride
  __shared__ float Mrow[32], Isum[32];
  __shared__ float Ypart[4][16][16];

  const float* qptr = q + (size_t)bh * L * D;
  const float* kptr = k + (size_t)bh * L * D;
  const float* vptr = v + (size_t)bh * L * D;

  {  // load 32x32 query tile (pre-scaled in QKV epilogue)
    const int r = tid >> 3, c4 = (tid & 7) * 4;
    *(float4*)&Qs[r][c4] = *(const float4*)(qptr + (size_t)(qbase + r) * D + c4);
  }
  __syncthreads();

  // ---- Pass 1: online row max / sum over Lk ----
  const int row = tid >> 3;         // 32 rows, 8 threads each
  const int sub = tid & 7;
  float rmax = -__builtin_inff(), rsum = 0.f;

  for (int kc = 0; kc < L; kc += 64) {
    score_chunk(Qs, S, kptr, bias, mask, b, h, qbase, kc, wave, lane);
    __syncthreads();

    float cmax = -__builtin_inff();
#pragma unroll
    for (int j = 0; j < 8; ++j) cmax = fmaxf(cmax, S[row][sub * 8 + j]);
    cmax = fmaxf(cmax, __shfl_xor(cmax, 1));
    cmax = fmaxf(cmax, __shfl_xor(cmax, 2));
    cmax = fmaxf(cmax, __shfl_xor(cmax, 4));
    const float nmax = fmaxf(rmax, cmax);
    float csum = 0.f;
#pragma unroll
    for (int j = 0; j < 8; ++j) csum += __expf(S[row][sub * 8 + j] - nmax);
    csum += __shfl_xor(csum, 1);
    csum += __shfl_xor(csum, 2);
    csum += __shfl_xor(csum, 4);
    rsum = rsum * __expf(rmax - nmax) + csum;
    rmax = nmax;
    __syncthreads();
  }
  if (sub == 0) { Mrow[row] = rmax; Isum[row] = 1.0f / rsum; }
  __syncthreads();

  // ---- Pass 2: recompute (k/bias now L2-hot), normalize, emit a + P@V ----
  const int qs2 = (wave >> 1) & 1;  // y subtile: query slice
  const int ds2 = wave & 1;         // y subtile: head-dim slice
  const int half = wave >> 2;       // k-split within 64-key chunk
  v8f yacc = {0.f, 0.f, 0.f, 0.f, 0.f, 0.f, 0.f, 0.f};

  for (int kc = 0; kc < L; kc += 64) {
    score_chunk(Qs, S, kptr, bias, mask, b, h, qbase, kc, wave, lane);
    __syncthreads();

#pragma unroll
    for (int j = 0; j < 8; ++j) {
      const int col = sub * 8 + j;
      S[row][col] = __expf(S[row][col] - Mrow[row]) * Isum[row];
    }
    __syncthreads();

    // write probabilities: a[b][lk][h][lq], lanes over contiguous lq (128B)
#pragma unroll
    for (int jj = 0; jj < 8; ++jj) {
      const int lkr = jj * 8 + wave;
      a_out[(((size_t)(b * L + kc + lkr)) * H + h) * L + qbase + lane] = S[lane][lkr];
    }

    // y += P @ V over this chunk's 64 keys (split across wave pairs)
    const int arow = qs2 * 16 + nl;
    const int dcol = ds2 * 16 + nl;
#pragma unroll
    for (int kk = 0; kk < 32; kk += 4) {
      const int ks = half * 32 + kk + 2 * hi;
      v2f af = *(const v2f*)&S[arow][ks];
      v2f bf = {vptr[(size_t)(kc + ks) * D + dcol],
                vptr[(size_t)(kc + ks + 1) * D + dcol]};
      yacc = WMMA_F32(af, bf, yacc);
    }
    __syncthreads();
  }

  // combine k-split halves, write y in bqhc layout
  if (wave >= 4) {
#pragma unroll
    for (int vv = 0; vv < 8; ++vv) Ypart[wave & 3][vv + 8 * hi][nl] = yacc[vv];
  }
  __syncthreads();
  if (wave < 4) {
#pragma unroll
    for (int vv = 0; vv < 8; ++vv) {
      const float val = yacc[vv] + Ypart[wave][vv + 8 * hi][nl];
      const int lq = qbase + qs2 * 16 + vv + 8 * hi;
      y_ws[(((size_t)(b * L + lq)) * H + h) * D + ds2 * 16 + nl] = val;
    }
  }
}

// ---------------------------------------------------------------------------
extern "C" void kernel_launch(void* const* d_in, const int* in_sizes, int n_in,
                              void* d_out, int out_size, void* d_ws, size_t ws_size,
                              hipStream_t stream) {
  using namespace esm;
  const float* x      = (const float*)d_in[0];
  const unsigned char* mask = (const unsigned char*)d_in[1];  // jnp.bool_: 1 byte
  const float* bias   = (const float*)d_in[2];
  const float* w_qkv  = (const float*)d_in[3];
  const float* w_o    = (const float*)d_in[4];
  const float* b_o    = (const float*)d_in[5];
  const float* w_g    = (const float*)d_in[6];
  const float* b_g    = (const float*)d_in[7];

  float* y_out = (float*)d_out;                       // [B,L,C]
  float* a_out = y_out + (size_t)B * L * C;           // [B,Lk,H,Lq]

  const size_t NBH = (size_t)B * H * L * D;           // 2M floats
  float* q_ws  = (float*)d_ws;
  float* k_ws  = q_ws + NBH;
  float* v_ws  = k_ws + NBH;
  float* y_ws  = v_ws + NBH;                          // [B,L,C] (bqhc)
  float* g_ws  = y_ws + (size_t)B * L * C;            // gated

  const int M = B * L;  // 2048

  // 1) QKV projection + head split + q scaling (TDM double-buffered)
  gemm_wmma_f32<0><<<dim3((3 * C) / 64, M / 64), 256, 0, stream>>>(
      x, w_qkv, nullptr, nullptr, q_ws, k_ws, v_ws, M, 3 * C, C);

  // 2) attention: softmax probs -> a_out, context -> y_ws
  attn_wmma_f32<<<dim3(H, L / 32, B), 256, 0, stream>>>(
      q_ws, k_ws, v_ws, bias, mask, a_out, y_ws);

  // 3) gate: sigmoid(x @ w_g^T + b_g) * y
  gemm_wmma_f32<1><<<dim3(C / 64, M / 64), 256, 0, stream>>>(
      x, w_g, b_g, y_ws, g_ws, nullptr, nullptr, M, C, C);

  // 4) output projection
  gemm_wmma_f32<2><<<dim3(C / 64, M / 64), 256, 0, stream>>>(
      g_ws, w_o, b_o, nullptr, y_out, nullptr, nullptr, M, C, C);
}